// CoxTimeDependentLoss_83880711291605
// MI455X (gfx1250) — compile-verified
//
#include <hip/hip_runtime.h>
#include <hip/hip_bf16.h>
#include <stdint.h>

// ---------------------------------------------------------------------------
// Cox time-dependent loss, MI455X (gfx1250, wave32).
// Memory-bound scan/histogram problem: ~150MB traffic -> ~10us at 23.3TB/s;
// working set (~110MB) fits the 192MB L2. Deterministic via integer (Q33
// fixed point) accumulation everywhere order can vary. Scan tiles are staged
// global->LDS with the CDNA5 async-LDS engine (global_load_async_to_lds_b128,
// ASYNCcnt + s_wait_asynccnt).
// ---------------------------------------------------------------------------

#define FIXSCALE 8589934592.0      // 2^33
#define INV_FIXSCALE (1.0 / 8589934592.0)

static constexpr int SCAN_BLOCK = 256;   // 8 wave32 waves per block
static constexpr int SCAN_TPT   = 16;
static constexpr int SCAN_TILE  = SCAN_BLOCK * SCAN_TPT; // 4096

#if defined(__has_builtin)
#if __has_builtin(__builtin_amdgcn_global_load_async_to_lds_b128)
#define HAVE_ASYNC_LDS 1
#endif
#endif

#define GAS __attribute__((address_space(1)))   // printed as __device__ by clang
#define LAS __attribute__((address_space(3)))   // printed as __shared__ by clang

typedef __attribute__((ext_vector_type(4))) int v4i;

__device__ __forceinline__ void wait_async0() {
#if defined(__has_builtin) && __has_builtin(__builtin_amdgcn_s_wait_asynccnt)
  __builtin_amdgcn_s_wait_asynccnt(0);
#else
  asm volatile("s_wait_asynccnt 0x0" ::: "memory");
#endif
}

__device__ __forceinline__ int bucket_of(float t, int nb, double scale) {
  double x = (double)t * scale;          // monotone in t; ties -> same bucket
  int b = (int)x;
  if (b < 0) b = 0;
  if (b > nb - 1) b = nb - 1;
  return b;
}

__device__ __forceinline__ unsigned long long efixed_of(float r) {
  double e = exp((double)r);
  // explicit fma so every kernel produces bit-identical fixed-point values
  return (unsigned long long)fma(e, FIXSCALE, 0.5);
}

// ---- pass 1: per-bucket count + per-bucket sum(exp(r)) in Q33 u64 ----------
__global__ void cox_histo(const float* __restrict__ risk,
                          const float* __restrict__ y,
                          unsigned* __restrict__ count,
                          unsigned long long* __restrict__ bsum,
                          int n, int nb, double scale) {
  int i = blockIdx.x * blockDim.x + threadIdx.x;
  if (i >= n) return;
  float t = y[2 * i];
  int b = bucket_of(t, nb, scale);
  atomicAdd(&count[b], 1u);
  atomicAdd(&bsum[b], efixed_of(risk[i]));
}

// ---- tile-level exclusive scan (exact: integer adds, order-independent) ----
// Async path: DMA the 4096-element tile into LDS (padded layout: per-thread
// chunk stride SCAN_TPT + 16/sizeof(T) elems => 16B-aligned chunks, <=2-way
// bank conflicts on the sequential per-thread reads).
template <typename T>
__global__ void scan_tile_kernel(const T* __restrict__ in, T* __restrict__ out,
                                 T* __restrict__ tileSums) {
  __shared__ T sdata[SCAN_BLOCK];
  const int t = threadIdx.x;
  const long long base = (long long)blockIdx.x * SCAN_TILE + (long long)t * SCAN_TPT;
  T v[SCAN_TPT], ex[SCAN_TPT];

#ifdef HAVE_ASYNC_LDS
  constexpr int ELEM16 = 16 / (int)sizeof(T);          // elems per 16B chunk
  constexpr int STRIDE = SCAN_TPT + ELEM16;            // padded chunk stride
  constexpr int C16PT  = SCAN_TPT / ELEM16;            // 16B chunks per thread
  __shared__ T tileLds[SCAN_BLOCK * STRIDE];
  const char* gbase = (const char*)(in + (size_t)blockIdx.x * SCAN_TILE);
#pragma unroll
  for (int r = 0; r < C16PT; r++) {                    // coalesced staging
    int c = r * SCAN_BLOCK + t;                        // 16B chunk id in tile
    int owner = c / C16PT, within = c % C16PT;
    size_t gByte = (size_t)c * 16;
    size_t lByte = (size_t)owner * STRIDE * sizeof(T) + (size_t)within * 16;
    __builtin_amdgcn_global_load_async_to_lds_b128(
        (GAS v4i*)(gbase + gByte),
        (LAS v4i*)((char*)tileLds + lByte), 0, 0);
  }
  wait_async0();        // this wave's async ops complete (LDS written)
  __syncthreads();      // order across waves before cross-thread consumption
#pragma unroll
  for (int k = 0; k < SCAN_TPT; k++) v[k] = tileLds[t * STRIDE + k];
#else
#pragma unroll
  for (int k = 0; k < SCAN_TPT; k++) v[k] = in[base + k];
#endif

  T s = (T)0;
#pragma unroll
  for (int k = 0; k < SCAN_TPT; k++) { ex[k] = s; s += v[k]; }
  sdata[t] = s;
  for (int off = 1; off < SCAN_BLOCK; off <<= 1) {
    __syncthreads();
    T x = (t >= off) ? sdata[t - off] : (T)0;
    __syncthreads();
    sdata[t] += x;
  }
  __syncthreads();
  T texcl = t ? sdata[t - 1] : (T)0;
  if (t == SCAN_BLOCK - 1) tileSums[blockIdx.x] = sdata[SCAN_BLOCK - 1];
#pragma unroll
  for (int k = 0; k < SCAN_TPT; k++) out[base + k] = ex[k] + texcl;
}

// ---- single-block parallel exclusive scan (n <= 2048) ----------------------
template <typename T>
__global__ void scan_block_kernel(T* data, int n) {
  __shared__ T sdata[SCAN_BLOCK];
  const int t = threadIdx.x;
  const int P = (n + SCAN_BLOCK - 1) / SCAN_BLOCK;     // <= 8
  T v[8];
  T s = (T)0;
  for (int k = 0; k < P; k++) {
    int i = t * P + k;
    v[k] = (i < n) ? data[i] : (T)0;
    s += v[k];
  }
  sdata[t] = s;
  for (int off = 1; off < SCAN_BLOCK; off <<= 1) {
    __syncthreads();
    T x = (t >= off) ? sdata[t - off] : (T)0;
    __syncthreads();
    sdata[t] += x;
  }
  __syncthreads();
  T run = t ? sdata[t - 1] : (T)0;
  for (int k = 0; k < P; k++) {
    int i = t * P + k;
    if (i < n) data[i] = run;
    run += v[k];
  }
}

template <typename T>
__global__ void scan_add_kernel(T* __restrict__ out,
                                const T* __restrict__ tileSums) {
  long long i = (long long)blockIdx.x * blockDim.x + threadIdx.x;
  out[i] += tileSums[i / SCAN_TILE];
}

// ---- pass 2: counting-sort scatter of (t, eFixed) into bucket segments -----
__global__ void cox_scatter(const float* __restrict__ risk,
                            const float* __restrict__ y,
                            unsigned* __restrict__ cursor,
                            float* __restrict__ sT,
                            unsigned long long* __restrict__ sE,
                            int n, int nb, double scale) {
  int i = blockIdx.x * blockDim.x + threadIdx.x;
  if (i >= n) return;
  float t = y[2 * i];
  int b = bucket_of(t, nb, scale);
  unsigned pos = atomicAdd(&cursor[b], 1u);
  sT[pos] = t;
  sE[pos] = efixed_of(risk[i]);
}

// ---- pass 3: risk set + partial; per-block exact Q33 sums (no atomics) -----
__global__ void cox_final(const float* __restrict__ risk,
                          const float* __restrict__ y,
                          const unsigned long long* __restrict__ prefix,
                          const unsigned long long* __restrict__ bsum,
                          const unsigned* __restrict__ offset,
                          const unsigned* __restrict__ count,
                          const float* __restrict__ sT,
                          const unsigned long long* __restrict__ sE,
                          long long* __restrict__ blockSums,
                          int n, int nb, double scale) {
  __shared__ long long red[SCAN_BLOCK];
  int i = blockIdx.x * blockDim.x + threadIdx.x;
  long long contrib = 0;
  if (i < n) {
    float ev = y[2 * i + 1];
    if (ev != 0.0f) {
      float t = y[2 * i];
      int b = bucket_of(t, nb, scale);
      unsigned beg = offset[b];
      unsigned end = beg + count[b];
      // gfx1250 global_prefetch_b8 for the within-bucket segment
      __builtin_prefetch(&sT[beg], 0, 1);
      __builtin_prefetch(&sE[beg], 0, 1);
      unsigned long long total = prefix[nb - 1] + bsum[nb - 1];
      unsigned long long within = 0;
      for (unsigned j = beg; j < end; j++)
        if (sT[j] < t) within += sE[j];       // exact tie handling ('left')
      unsigned long long riskF = total - (prefix[b] + within);
      double riskv = (double)riskF * INV_FIXSCALE;   // > 0 always (i in set)
      double e = exp((double)risk[i]);
      double partial = (double)ev * (e - log(riskv));
      double sc = partial * FIXSCALE;
      contrib = (long long)(sc + (sc >= 0.0 ? 0.5 : -0.5));
    }
  }
  red[threadIdx.x] = contrib;
  __syncthreads();
  for (int off = SCAN_BLOCK / 2; off > 0; off >>= 1) {
    if (threadIdx.x < off) red[threadIdx.x] += red[threadIdx.x + off];
    __syncthreads();
  }
  if (threadIdx.x == 0) blockSums[blockIdx.x] = red[0];
}

// ---- final deterministic integer reduction of per-block sums ---------------
__global__ void cox_reduce(const long long* __restrict__ blockSums,
                           int nblocks, float* __restrict__ out) {
  __shared__ long long red[SCAN_BLOCK];
  long long s = 0;
  for (int i = threadIdx.x; i < nblocks; i += SCAN_BLOCK) s += blockSums[i];
  red[threadIdx.x] = s;
  __syncthreads();
  for (int off = SCAN_BLOCK / 2; off > 0; off >>= 1) {
    if (threadIdx.x < off) red[threadIdx.x] += red[threadIdx.x + off];
    __syncthreads();
  }
  if (threadIdx.x == 0)
    out[0] = (float)(-(double)red[0] * INV_FIXSCALE);
}

// ---------------------------------------------------------------------------
extern "C" void kernel_launch(void* const* d_in, const int* in_sizes, int n_in,
                              void* d_out, int out_size, void* d_ws,
                              size_t ws_size, hipStream_t stream) {
  const float* risk = (const float*)d_in[0];
  const float* y    = (const float*)d_in[1];   // [N,2] row-major: (time,event)
  float* out        = (float*)d_out;
  const int N = in_sizes[0];

  const int eblk = 256;
  const int egrd = (N + eblk - 1) / eblk;

  // Pick bucket count: avg ~2 elems/bucket at 2^21, shrink if ws is small.
  int nb = 1 << 21;
  while (nb > (1 << 16)) {
    size_t req = (size_t)nb * 28 + ((size_t)nb / SCAN_TILE) * 12 +
                 (size_t)N * 12 + (size_t)egrd * 8 + (size_t)16 * 4096;
    if (req <= ws_size) break;
    nb >>= 1;
  }
  const int tiles = nb / SCAN_TILE;
  const double scale = (double)nb / 1000.0;

  // Carve workspace (256B aligned slices).
  uint8_t* p = (uint8_t*)d_ws;
  auto carve = [&](size_t bytes) -> void* {
    void* r = (void*)p;
    p += (bytes + 255) & ~(size_t)255;
    return r;
  };
  unsigned long long* bsum   = (unsigned long long*)carve((size_t)nb * 8);
  unsigned long long* prefix = (unsigned long long*)carve((size_t)nb * 8);
  unsigned long long* ts64   = (unsigned long long*)carve((size_t)tiles * 8);
  unsigned*           count  = (unsigned*)carve((size_t)nb * 4);
  unsigned*           offset = (unsigned*)carve((size_t)nb * 4);
  unsigned*           cursor = (unsigned*)carve((size_t)nb * 4);
  unsigned*           ts32   = (unsigned*)carve((size_t)tiles * 4);
  long long*          bsums  = (long long*)carve((size_t)egrd * 8);
  float*              sT     = (float*)carve((size_t)N * 4);
  unsigned long long* sE     = (unsigned long long*)carve((size_t)N * 8);

  // Zero the atomically-accumulated buffers every call (capture-safe).
  (void)hipMemsetAsync(bsum, 0, (size_t)nb * 8, stream);
  (void)hipMemsetAsync(count, 0, (size_t)nb * 4, stream);

  // 1) bucket histogram: counts + Q33 exp-sums
  cox_histo<<<egrd, eblk, 0, stream>>>(risk, y, count, bsum, N, nb, scale);

  // 2a) exclusive scan of bucket sums (u64)
  scan_tile_kernel<unsigned long long><<<tiles, SCAN_BLOCK, 0, stream>>>(bsum, prefix, ts64);
  scan_block_kernel<unsigned long long><<<1, SCAN_BLOCK, 0, stream>>>(ts64, tiles);
  scan_add_kernel<unsigned long long><<<nb / SCAN_BLOCK, SCAN_BLOCK, 0, stream>>>(prefix, ts64);

  // 2b) exclusive scan of counts (u32) -> bucket offsets
  scan_tile_kernel<unsigned><<<tiles, SCAN_BLOCK, 0, stream>>>(count, offset, ts32);
  scan_block_kernel<unsigned><<<1, SCAN_BLOCK, 0, stream>>>(ts32, tiles);
  scan_add_kernel<unsigned><<<nb / SCAN_BLOCK, SCAN_BLOCK, 0, stream>>>(offset, ts32);

  // cursor = offset copy for scatter tickets
  (void)hipMemcpyAsync(cursor, offset, (size_t)nb * 4,
                       hipMemcpyDeviceToDevice, stream);

  // 3) counting-sort scatter
  cox_scatter<<<egrd, eblk, 0, stream>>>(risk, y, cursor, sT, sE, N, nb, scale);

  // 4) risk sets + per-block exact partial sums
  cox_final<<<egrd, eblk, 0, stream>>>(risk, y, prefix, bsum, offset, count,
                                       sT, sE, bsums, N, nb, scale);

  // 5) deterministic integer reduction -> scalar loss
  cox_reduce<<<1, SCAN_BLOCK, 0, stream>>>(bsums, egrd, out);
}